// CIF_7834020348487
// MI455X (gfx1250) — compile-verified
//
#include <hip/hip_runtime.h>

#define N_MAX 384
#define THRESH 1.0f
#define EPS_TAIL 1e-4f
#define CHUNK 128
#define CHUNKP (CHUNK + 4)
#define NGRP (CHUNK / 4)
#define SCAN_CHUNK 2048
#define TOK_SENTINEL (-0x40000000)

typedef __attribute__((ext_vector_type(2))) float v2f;
typedef __attribute__((ext_vector_type(8))) float v8f;

// Packed per-frame record: token index + weight into token + weight into token+1.
struct alignas(16) Rec { int tok; float w0; float w1; int pad; };

// ---------------------------------------------------------------------------
// Phase 1: per-batch alpha scaling + sequential integrate-and-fire scan.
// ---------------------------------------------------------------------------
__global__ __launch_bounds__(256)
void cif_scan_kernel(const float* __restrict__ alpha,
                     const int*   __restrict__ target_len,
                     Rec*   __restrict__ recA,
                     int*   __restrict__ tokfirstA,
                     int*   __restrict__ nvalidA,
                     int*   __restrict__ token_lens,
                     int*   __restrict__ boundsA,
                     int T)
{
    const int b   = blockIdx.x;
    const int tid = threadIdx.x;

    __shared__ float s_red[256];
    __shared__ float s_a[SCAN_CHUNK];

    const float* al = alpha + (size_t)b * T;

    // ---- sum(alpha) reduction ----
    float s = 0.f;
    for (int t = tid; t < T; t += 256) s += al[t];
    s_red[tid] = s;
    __syncthreads();
    for (int off = 128; off > 0; off >>= 1) {
        if (tid < off) s_red[tid] += s_red[tid + off];
        __syncthreads();
    }
    const float scale = (float)target_len[b] / (s_red[0] + 1e-6f);

    // ---- init tokfirst = T everywhere, bounds = 0 ----
    int* tf  = tokfirstA + (size_t)b * (N_MAX + 17);
    for (int i = tid; i < N_MAX + 17; i += 256) tf[i] = T;
    int* bnd = boundsA + (size_t)b * N_MAX * 2;
    for (int i = tid; i < N_MAX * 2; i += 256) bnd[i] = 0;
    __syncthreads();

    // ---- sequential scan: thread 0 carries state across LDS-staged chunks ----
    float integ = 0.f;
    int   ntok  = 0;
    int   ts    = -1;
    Rec*  rp = recA + (size_t)b * T;

    for (int base = 0; base < T; base += SCAN_CHUNK) {
        int cnt = T - base; if (cnt > SCAN_CHUNK) cnt = SCAN_CHUNK;
        for (int i = tid; i < cnt; i += 256) s_a[i] = al[base + i] * scale;
        __syncthreads();
        if (tid == 0) {
            if (base == 0) tf[0] = 0;
            for (int i = 0; i < cnt; ++i) {
                const int   t = base + i;
                const float a = s_a[i];
                if (ts < 0 && a > 0.f) ts = t;             // token_start
                const float in_new      = integ + a;
                const bool  fire        = in_new >= THRESH;
                const float remain      = THRESH - integ;
                const float integ_after = fire ? (in_new - THRESH) : in_new;
                Rec r;
                r.tok = ntok;
                r.pad = 0;
                if (fire) {
                    r.w0 = remain;                         // emitted into token ntok
                    r.w1 = integ_after;                    // seeds token ntok+1
                    if (ntok < N_MAX) { bnd[ntok * 2] = ts; bnd[ntok * 2 + 1] = t + 1; }
                    ntok++;
                    if (ntok < N_MAX + 17) tf[ntok] = t + 1;
                    ts = (integ_after > 0.f) ? t : -1;
                } else {
                    r.w0 = a;
                    r.w1 = 0.f;
                }
                rp[t] = r;                                 // single 16B store
                integ = integ_after;
            }
        }
        __syncthreads();
    }

    if (tid == 0) {
        const bool tail = integ > EPS_TAIL;
        if (tail && ntok < N_MAX) { bnd[ntok * 2] = ts; bnd[ntok * 2 + 1] = T; }
        int nt = ntok + (tail ? 1 : 0);
        if (nt > N_MAX) nt = N_MAX;
        token_lens[b] = nt;
        nvalidA[b]    = nt;
    }
}

// ---------------------------------------------------------------------------
// Helper: shared A-operand value for one (frame record, token row) pair.
// ---------------------------------------------------------------------------
__device__ __forceinline__ float a_weight(const Rec& r, int row) {
    float a = 0.f;
    a = (r.tok + 1 == row) ? r.w1 : a;
    a = (r.tok     == row) ? r.w0 : a;
    return a;
}

// ---------------------------------------------------------------------------
// Phase 2: banded C = W x H via V_WMMA_F32_16X16X4_F32 (fp32 in, fp32 acc).
// Block = 256 threads = 8 waves; each wave owns a 16-dim slice of a
// 16-token x 128-dim output tile. A tiles are precomputed once per block into
// LDS (shared by all 8 waves); hot loop = ds_b64 + 2x global_b32 + wmma.
// DC > 0: compile-time embedding dim (enables immediate-offset B addressing).
// ---------------------------------------------------------------------------
template<int DC>
__global__ __launch_bounds__(256)
void cif_wmma_kernel(const float* __restrict__ h,
                     const Rec*   __restrict__ recA,
                     const int*   __restrict__ tokfirstA,
                     const int*   __restrict__ nvalidA,
                     float* __restrict__ out,
                     int T, int Drt)
{
    const int D    = (DC > 0) ? DC : Drt;                // compile-time if DC>0
    const int b    = blockIdx.z;
    const int n0   = blockIdx.x * 16;
    const int wave = threadIdx.x >> 5;
    const int lane = threadIdx.x & 31;
    const int half = lane >> 4;
    const int l16  = lane & 15;
    const int dw   = blockIdx.y * 128 + (wave << 4);     // wave's 16-dim slice

    __shared__ Rec s_rec[CHUNKP];
    __shared__ v2f s_A[NGRP * 32];                       // A tiles for whole chunk

    const int* tf = tokfirstA + (size_t)b * (N_MAX + 17);
    int t_lo = tf[n0] - 1; if (t_lo < 0) t_lo = 0;       // fire frame of token n0-1
    int hi   = n0 + 16;    if (hi > N_MAX) hi = N_MAX;
    int t_hi = tf[hi];     if (t_hi > T) t_hi = T;

    int       col  = dw + l16;                           // B-matrix col (dim)
    const int dstc = col;                                // store column (guarded)
    if (col > D - 1) col = D - 1;                        // clamp: load always legal

    const Rec*   rp     = recA + (size_t)b * T;
    const float* hbase0 = h + (size_t)b * T * D + col;   // this lane's column

    v8f acc = {};
    int tc = t_lo;

    // ================= full chunks: no clamps, no sentinels =================
    for (; tc + CHUNK <= t_hi; tc += CHUNK) {
        for (int i = threadIdx.x; i < CHUNK; i += 256)
            s_rec[i] = rp[tc + i];                       // b128 global -> ds
        __syncthreads();

        // cooperative A-tile build: NGRP groups x 32 lane-slots, built once
        for (int idx = threadIdx.x; idx < NGRP * 32; idx += 256) {
            const int g  = idx >> 5;
            const int L  = idx & 31;
            const int ki = (g << 2) + ((L >> 4) << 1);   // K-slot: 2*half + j
            const int rr = n0 + (L & 15);
            v2f a;
            a[0] = a_weight(s_rec[ki],     rr);
            a[1] = a_weight(s_rec[ki + 1], rr);
            s_A[idx] = a;
        }
        if (tc + CHUNK < t_hi)
            __builtin_prefetch(hbase0 + (size_t)(tc + CHUNK + (threadIdx.x >> 4)) * D, 0, 1);
        __syncthreads();

        // chunk-local base pointer: loads index with small (imm-foldable) offsets
        const float* pb = hbase0 + (size_t)(tc + (half << 1)) * D;
        const v2f*   pa = s_A + lane;
#pragma unroll 4
        for (int g4 = 0; g4 < NGRP; ++g4) {
            v2f A = pa[g4 << 5];                         // ds_load_b64
            v2f Bv;
            Bv[0] = pb[(g4 * 4) * D];                    // const offsets when DC>0
            Bv[1] = pb[(g4 * 4 + 1) * D];
            acc = __builtin_amdgcn_wmma_f32_16x16x4_f32(
                false, A, false, Bv, (short)0, acc, false, false);
        }
        __syncthreads();
    }

    // ================= tail partial chunk: clamped =================
    if (tc < t_hi) {
        const int cnt = t_hi - tc;
        for (int i = threadIdx.x; i < cnt; i += 256)
            s_rec[i] = rp[tc + i];
        if (threadIdx.x < 4) {                           // sentinel pad
            Rec z; z.tok = TOK_SENTINEL; z.w0 = 0.f; z.w1 = 0.f; z.pad = 0;
            s_rec[cnt + threadIdx.x] = z;
        }
        __syncthreads();

        const int ng = (cnt + 3) >> 2;
        for (int idx = threadIdx.x; idx < (ng << 5); idx += 256) {
            const int g  = idx >> 5;
            const int L  = idx & 31;
            const int ki = (g << 2) + ((L >> 4) << 1);
            const int rr = n0 + (L & 15);
            v2f a;
            a[0] = a_weight(s_rec[ki],     rr);
            a[1] = a_weight(s_rec[ki + 1], rr);
            s_A[idx] = a;
        }
        __syncthreads();

        const int kmax = (T - 1) - tc;                   // uniform clamp bound
        for (int g = 0; g < cnt; g += 4) {
            v2f A = s_A[((g >> 2) << 5) + lane];
            int k0 = g + (half << 1);     if (k0 > kmax) k0 = kmax;
            int k1 = g + (half << 1) + 1; if (k1 > kmax) k1 = kmax;
            v2f Bv;
            Bv[0] = hbase0[(size_t)(tc + k0) * D];
            Bv[1] = hbase0[(size_t)(tc + k1) * D];
            acc = __builtin_amdgcn_wmma_f32_16x16x4_f32(
                false, A, false, Bv, (short)0, acc, false, false);
        }
    }

    // C/D layout: VGPR r, lane-half hf -> token row M = r + 8*hf, col N = lane&15
    const int nv = nvalidA[b];
#pragma unroll
    for (int r = 0; r < 8; ++r) {
        const int n = n0 + r + (half << 3);
        if (dstc < D) {
            float v = (n < nv) ? acc[r] : 0.f;           // drop un-emitted tokens
            out[((size_t)b * N_MAX + n) * D + dstc] = v;
        }
    }
}

// ---------------------------------------------------------------------------
extern "C" void kernel_launch(void* const* d_in, const int* in_sizes, int n_in,
                              void* d_out, int out_size, void* d_ws, size_t ws_size,
                              hipStream_t stream)
{
    const float* h          = (const float*)d_in[0];
    const float* alpha      = (const float*)d_in[1];
    const int*   target_len = (const int*)d_in[2];

    const int B = in_sizes[2];
    const int T = in_sizes[1] / B;
    const int D = in_sizes[0] / (B * T);

    float* out_c      = (float*)d_out;
    int*   token_lens = (int*)(out_c + (size_t)B * N_MAX * D);
    int*   bounds     = token_lens + B;

    // workspace carve-up
    char* ws = (char*)d_ws;
    Rec*  rec      = (Rec*)ws;    ws += sizeof(Rec) * (size_t)B * T;
    int*  tokfirst = (int*)ws;    ws += sizeof(int) * (size_t)B * (N_MAX + 17);
    int*  nvalid   = (int*)ws;

    cif_scan_kernel<<<B, 256, 0, stream>>>(alpha, target_len, rec,
                                           tokfirst, nvalid, token_lens, bounds, T);

    dim3 grid(N_MAX / 16, (D + 127) / 128, B);
    if (D == 768) {
        cif_wmma_kernel<768><<<grid, 256, 0, stream>>>(h, rec, tokfirst, nvalid,
                                                       out_c, T, D);
    } else {
        cif_wmma_kernel<0><<<grid, 256, 0, stream>>>(h, rec, tokfirst, nvalid,
                                                     out_c, T, D);
    }
}